// LSTMPredictor_68367289418345
// MI455X (gfx1250) — compile-verified
//
#include <hip/hip_runtime.h>
#include <hip/hip_bf16.h>

// ---------------------------------------------------------------------------
// LSTMPredictor on MI455X (gfx1250, wave32, WMMA).
// fp32 path via V_WMMA_F32_16X16X4_F32 to preserve accuracy across the
// 600-step recurrence. Big input-projection GEMMs are parallel; recurrences
// are persistent kernels with a device-wide barrier per timestep and the
// LSTM cell state kept entirely in VGPRs.
// ---------------------------------------------------------------------------

typedef float v2f __attribute__((ext_vector_type(2)));
typedef float v8f __attribute__((ext_vector_type(8)));

#define WMMA_F32(a, b, c) \
  __builtin_amdgcn_wmma_f32_16x16x4_f32(false, (a), false, (b), (short)0, (c), false, false)

__device__ __forceinline__ float sigf(float x) { return 1.0f / (1.0f + __expf(-x)); }

// ---------------------------------------------------------------------------
// Device-wide sense barrier (agent-scope atomics in global scratch).
// sync[0] = arrival counter, sync[32] = generation (separate cache lines).
// ---------------------------------------------------------------------------
__device__ __forceinline__ void grid_sync(unsigned* sync, unsigned nwg) {
  __threadfence();
  __syncthreads();
  if (threadIdx.x == 0) {
    unsigned* cnt = sync;
    unsigned* gen = sync + 32;
    unsigned g = __hip_atomic_load(gen, __ATOMIC_RELAXED, __HIP_MEMORY_SCOPE_AGENT);
    unsigned a = __hip_atomic_fetch_add(cnt, 1u, __ATOMIC_ACQ_REL, __HIP_MEMORY_SCOPE_AGENT);
    if (a == nwg - 1u) {
      __hip_atomic_store(cnt, 0u, __ATOMIC_RELAXED, __HIP_MEMORY_SCOPE_AGENT);
      __hip_atomic_fetch_add(gen, 1u, __ATOMIC_RELEASE, __HIP_MEMORY_SCOPE_AGENT);
    } else {
      while (__hip_atomic_load(gen, __ATOMIC_ACQUIRE, __HIP_MEMORY_SCOPE_AGENT) == g) {
        __builtin_amdgcn_s_sleep(8);
      }
    }
  }
  __syncthreads();
  __threadfence();
}

// ---------------------------------------------------------------------------
// Generic GEMM:  Y[M,N] = concat(A1[M,K1], A2[M,K2]) @ W[N,K]^T + b1 (+ b2)
// M = 76800 (multiple of 128). Block tile 128x64, 8 waves, each wave 32x32
// (2x2 WMMA f32 16x16x4 accumulators). K staged through LDS in 32-chunks.
// ---------------------------------------------------------------------------
__global__ __launch_bounds__(256) void gemm_bias_kernel(
    const float* __restrict__ A1, int K1, const float* __restrict__ A2, int K2,
    const float* __restrict__ W, const float* __restrict__ b1,
    const float* __restrict__ b2, float* __restrict__ Y, int N) {
  constexpr int KC = 32;
  __shared__ __align__(16) float As[128][KC + 4];  // row stride 36 floats = 144B (16B aligned)
  __shared__ __align__(16) float Ws[64][KC + 4];

  const int tid = threadIdx.x;
  const int lane = tid & 31;
  const int wave = tid >> 5;
  const int waveM = wave & 3;   // rows 32*waveM
  const int waveN = wave >> 2;  // cols 32*waveN
  const int half = lane >> 4;   // K-half selector for WMMA fragments
  const int l16 = lane & 15;
  const long rowBase = (long)blockIdx.y * 128;
  const int colBase = blockIdx.x * 64;
  const int K = K1 + K2;

  v8f acc[2][2];
  for (int i = 0; i < 2; ++i)
    for (int j = 0; j < 2; ++j)
      acc[i][j] = (v8f){0.f, 0.f, 0.f, 0.f, 0.f, 0.f, 0.f, 0.f};

  const int nChunks = K / KC;  // K1, K2 are multiples of 32
  for (int ch = 0; ch < nChunks; ++ch) {
    const int k0 = ch * KC;
    __syncthreads();
    // --- stage A chunk: 128 rows x 32 cols (float4, 4 per thread) ---
    {
      const float* src;
      int srcK, srcOff;
      if (k0 < K1) { src = A1; srcK = K1; srcOff = k0; }
      else         { src = A2; srcK = K2; srcOff = k0 - K1; }
      for (int it = 0; it < 4; ++it) {
        int idx = tid + it * 256;   // 0..1023
        int r = idx >> 3;
        int c4 = idx & 7;
        float4 v = *(const float4*)(src + (rowBase + r) * (long)srcK + srcOff + c4 * 4);
        *(float4*)(&As[r][c4 * 4]) = v;
      }
    }
    // --- stage W chunk: 64 rows x 32 cols (float4, 2 per thread) ---
    for (int it = 0; it < 2; ++it) {
      int idx = tid + it * 256;  // 0..511
      int r = idx >> 3;
      int c4 = idx & 7;
      int n = colBase + r;
      float4 v = make_float4(0.f, 0.f, 0.f, 0.f);
      if (n < N) v = *(const float4*)(W + (long)n * K + k0 + c4 * 4);
      *(float4*)(&Ws[r][c4 * 4]) = v;
    }
    __syncthreads();
    // --- 8 WMMA k-steps over this chunk ---
    for (int ks = 0; ks < KC; ks += 4) {
      const int kk = ks + half * 2;
      v2f a0 = *(const v2f*)(&As[32 * waveM + l16][kk]);
      v2f a1 = *(const v2f*)(&As[32 * waveM + 16 + l16][kk]);
      v2f b0 = *(const v2f*)(&Ws[32 * waveN + l16][kk]);
      v2f b1v = *(const v2f*)(&Ws[32 * waveN + 16 + l16][kk]);
      acc[0][0] = WMMA_F32(a0, b0, acc[0][0]);
      acc[0][1] = WMMA_F32(a0, b1v, acc[0][1]);
      acc[1][0] = WMMA_F32(a1, b0, acc[1][0]);
      acc[1][1] = WMMA_F32(a1, b1v, acc[1][1]);
    }
  }

  // --- epilogue: add bias, store (C/D layout: vgpr r -> m = r + 8*half) ---
  for (int i = 0; i < 2; ++i)
    for (int j = 0; j < 2; ++j) {
      int colT = colBase + 32 * waveN + 16 * j + l16;
      if (colT < N) {
        float bias = b1[colT] + (b2 ? b2[colT] : 0.f);
        for (int r = 0; r < 8; ++r) {
          long row = rowBase + 32 * waveM + 16 * i + r + 8 * half;
          Y[row * (long)N + colT] = acc[i][j][r] + bias;
        }
      }
    }
}

// ---------------------------------------------------------------------------
// Persistent LSTM recurrence. Grid = H/16 workgroups, 256 threads (8 waves).
// Workgroup b owns output columns [16b,16b+16) in ALL FOUR gate groups, so
// the elementwise stage is local (no cross-WG gather). Wave w owns rows
// [16w,16w+16). Cell state c: one v8f per wave, lives in VGPRs for all steps.
// Per step: acc[g] seeded from precomputed xg (biases folded), K=H WMMA
// reduction with h_prev and the WG's 64 Whh rows staged through LDS, then
// gates -> c,h, h written to h_seq and the shared h_prev buffer, grid barrier.
// ---------------------------------------------------------------------------
template <int H, int S>
__global__ __launch_bounds__(256) void lstm_rec_kernel(
    const float* __restrict__ xg,    // [S, 128, 4H]  (biases folded in)
    const float* __restrict__ Whh,   // [4H, H]
    float* __restrict__ hseq,        // [S, 128, H]
    float* __restrict__ hprev,       // [128, H] scratch
    unsigned* __restrict__ sync) {
  constexpr int G4 = 4 * H;
  constexpr int CH = 64;          // K chunk width
  constexpr unsigned NWG = H / 16;
  __shared__ __align__(16) float hs[128][CH + 4];  // 34.8 KB
  __shared__ __align__(16) float ws[64][CH + 4];   // 17.4 KB

  const int tid = threadIdx.x;
  const int lane = tid & 31;
  const int wave = tid >> 5;
  const int half = lane >> 4;
  const int l16 = lane & 15;
  const int jBase = blockIdx.x * 16;  // column slice within each gate group
  const int rowW = 16 * wave;         // wave's 16 batch rows

  // zero our slice of h_prev (workspace is poisoned), then device barrier
  for (int it = tid; it < 128 * 16; it += 256) {
    int r = it >> 4, c = it & 15;
    hprev[r * H + jBase + c] = 0.f;
  }
  grid_sync(sync, NWG);

  v8f c_acc = (v8f){0.f, 0.f, 0.f, 0.f, 0.f, 0.f, 0.f, 0.f};

  for (int t = 0; t < S; ++t) {
    // seed accumulators with xg tile (i,f,g,o slices for our columns)
    const float* xgt = xg + (long)t * (128 * (long)G4);
    v8f acc[4];
    for (int g = 0; g < 4; ++g)
      for (int r = 0; r < 8; ++r)
        acc[g][r] = xgt[(long)(rowW + r + 8 * half) * G4 + g * H + jBase + l16];

    // gates += h_prev @ Whh^T  (our 64 rows of Whh)
    for (int kc = 0; kc < H; kc += CH) {
      __syncthreads();
      // h_prev chunk: 128 x 64 floats -> 8 float4 per thread
      for (int it = 0; it < 8; ++it) {
        int idx = tid + it * 256;
        int r = idx >> 4;
        int c4 = idx & 15;
        *(float4*)(&hs[r][c4 * 4]) = *(const float4*)(hprev + r * H + kc + c4 * 4);
      }
      // weight chunk: 64 rows (4 gate groups x 16 cols) x 64 floats -> 4 per thread
      for (int it = 0; it < 4; ++it) {
        int idx = tid + it * 256;
        int r = idx >> 4;
        int c4 = idx & 15;
        int g = r >> 4;
        int n = g * H + jBase + (r & 15);
        *(float4*)(&ws[r][c4 * 4]) = *(const float4*)(Whh + (long)n * H + kc + c4 * 4);
      }
      __syncthreads();
      for (int ks = 0; ks < CH; ks += 4) {
        const int kk = ks + half * 2;
        v2f a = *(const v2f*)(&hs[rowW + l16][kk]);
        v2f b0 = *(const v2f*)(&ws[0 * 16 + l16][kk]);
        v2f b1v = *(const v2f*)(&ws[1 * 16 + l16][kk]);
        v2f b2v = *(const v2f*)(&ws[2 * 16 + l16][kk]);
        v2f b3 = *(const v2f*)(&ws[3 * 16 + l16][kk]);
        acc[0] = WMMA_F32(a, b0, acc[0]);
        acc[1] = WMMA_F32(a, b1v, acc[1]);
        acc[2] = WMMA_F32(a, b2v, acc[2]);
        acc[3] = WMMA_F32(a, b3, acc[3]);
      }
    }

    // elementwise LSTM cell; accumulator (vgpr r, lane) = (row rowW+r+8*half, col jBase+l16)
    float* hst = hseq + (long)t * (128 * (long)H);
    for (int r = 0; r < 8; ++r) {
      float iv = sigf(acc[0][r]);
      float fv = sigf(acc[1][r]);
      float gv = tanhf(acc[2][r]);
      float ov = sigf(acc[3][r]);
      float cv = fv * c_acc[r] + iv * gv;
      c_acc[r] = cv;
      float hv = ov * tanhf(cv);
      int row = rowW + r + 8 * half;
      int col = jBase + l16;
      hst[(long)row * H + col] = hv;
      hprev[row * H + col] = hv;
    }
    grid_sync(sync, NWG);  // publish h before anyone starts step t+1
  }
}

// ---------------------------------------------------------------------------
// Host: 7 launches on `stream` (graph-capture safe; only hipMemsetAsync +
// kernel launches). Workspace needed ~1.29 GiB.
// ---------------------------------------------------------------------------
extern "C" void kernel_launch(void* const* d_in, const int* in_sizes, int n_in,
                              void* d_out, int out_size, void* d_ws, size_t ws_size,
                              hipStream_t stream) {
  const float* x_a   = (const float*)d_in[0];
  const float* x_v   = (const float*)d_in[1];
  const float* Wih_a = (const float*)d_in[2];
  const float* Whh_a = (const float*)d_in[3];
  const float* bih_a = (const float*)d_in[4];
  const float* bhh_a = (const float*)d_in[5];
  const float* Wih_v = (const float*)d_in[6];
  const float* Whh_v = (const float*)d_in[7];
  const float* bih_v = (const float*)d_in[8];
  const float* bhh_v = (const float*)d_in[9];
  const float* Wih_m = (const float*)d_in[10];
  const float* Whh_m = (const float*)d_in[11];
  const float* bih_m = (const float*)d_in[12];
  const float* bhh_m = (const float*)d_in[13];
  const float* W_out = (const float*)d_in[14];
  const float* b_out = (const float*)d_in[15];
  float* out = (float*)d_out;

  const size_t M = 76800;  // S*B
  float* ws = (float*)d_ws;
  float* XGA = ws;                   // M*1024
  float* XGV = XGA + M * 1024;       // M*512
  float* XGM = XGV + M * 512;        // M*2048
  float* HAs = XGM + M * 2048;       // M*256
  float* HVs = HAs + M * 256;        // M*128
  float* HMs = HVs + M * 128;        // M*512
  float* HPREV = HMs + M * 512;      // 128*512
  unsigned* SYNC = (unsigned*)(HPREV + 128 * 512);  // 64 u32

  hipMemsetAsync(SYNC, 0, 256, stream);

  dim3 blk(256);
  // input projections (biases bih+bhh folded)
  gemm_bias_kernel<<<dim3(1024 / 64, 600), blk, 0, stream>>>(
      x_a, 128, nullptr, 0, Wih_a, bih_a, bhh_a, XGA, 1024);
  gemm_bias_kernel<<<dim3(512 / 64, 600), blk, 0, stream>>>(
      x_v, 64, nullptr, 0, Wih_v, bih_v, bhh_v, XGV, 512);
  // acoustic + visual recurrences
  lstm_rec_kernel<256, 600><<<dim3(16), blk, 0, stream>>>(XGA, Whh_a, HAs, HPREV, SYNC);
  lstm_rec_kernel<128, 600><<<dim3(8), blk, 0, stream>>>(XGV, Whh_v, HVs, HPREV, SYNC);
  // fused master input projection: A = [h_a | h_v] per row (K1=256, K2=128)
  gemm_bias_kernel<<<dim3(2048 / 64, 600), blk, 0, stream>>>(
      HAs, 256, HVs, 128, Wih_m, bih_m, bhh_m, XGM, 2048);
  // master recurrence
  lstm_rec_kernel<512, 600><<<dim3(32), blk, 0, stream>>>(XGM, Whh_m, HMs, HPREV, SYNC);
  // output projection (N=60, guarded)
  gemm_bias_kernel<<<dim3(1, 600), blk, 0, stream>>>(
      HMs, 512, nullptr, 0, W_out, b_out, nullptr, out, 60);
}